// Transforming_56195352101061
// MI455X (gfx1250) — compile-verified
//
#include <hip/hip_runtime.h>

typedef float v2f __attribute__((ext_vector_type(2)));
typedef float v8f __attribute__((ext_vector_type(8)));

#define B_SZ   64
#define N_IN   1152
#define N_OUT  10
#define D_IN   8
#define D_OUT  16

// votes[b,n,m,o] = sum_i inputs[b,n,i] * W[n,m,i,o]
// Per capsule n: (64x8) @ (8x160) GEMM done with V_WMMA_F32_16X16X4_F32,
// K=8 as two chained k=4 WMMAs. One wave = one 16-row batch tile x all 10 m.
__global__ __launch_bounds__(256) void caps_transform_wmma(
    const float* __restrict__ in,   // [64,1152,8]
    const float* __restrict__ w,    // [1152,10,8,16]
    float* __restrict__ out)        // [64,1152,10,16]
{
  const int lane   = threadIdx.x & 31;
  const int waveId = blockIdx.x * 8 + (threadIdx.x >> 5);  // 0..4607
  const int n      = waveId >> 2;    // capsule index 0..1151
  const int mtile  = waveId & 3;     // batch tile: rows 16*mtile .. 16*mtile+15

  const int half = lane >> 4;        // 0: lanes 0-15 (K 0,1), 1: lanes 16-31 (K 2,3)
  const int l16  = lane & 15;

  // ---- A fragments: inputs[b, n, k]; 16x8 tile as two 16x4 f32 fragments
  const int brow = mtile * 16 + l16;
  const float* aPtr = in + ((size_t)brow * N_IN + n) * D_IN + half * 2;
  v2f a0 = *(const v2f*)(aPtr);      // K = {0,1} | {2,3}
  v2f a1 = *(const v2f*)(aPtr + 4);  // K = {4,5} | {6,7}

  const float* wBaseN = w + (size_t)n * (N_OUT * D_IN * D_OUT);
  const int kb = half * 2;

  for (int m = 0; m < N_OUT; ++m) {
    const float* wm = wBaseN + m * (D_IN * D_OUT);
    // B 4x16 fragments: lane holds col N=l16; VGPR0/1 hold rows K, K+1
    v2f b0 = { wm[(kb + 0) * D_OUT + l16], wm[(kb + 1) * D_OUT + l16] };
    v2f b1 = { wm[(kb + 4) * D_OUT + l16], wm[(kb + 5) * D_OUT + l16] };

    v8f c = {0.f, 0.f, 0.f, 0.f, 0.f, 0.f, 0.f, 0.f};
    // 8 args: (neg_a, A, neg_b, B, c_mod, C, reuse_a, reuse_b)
    c = __builtin_amdgcn_wmma_f32_16x16x4_f32(false, a0, false, b0,
                                              (short)0, c, false, false);
    c = __builtin_amdgcn_wmma_f32_16x16x4_f32(false, a1, false, b1,
                                              (short)0, c, false, false);

    // C/D layout: VGPR v -> row M = v + half*8, col = l16.
    // Lanes 0-15 of each VGPR store 64 contiguous bytes -> coalesced stream.
    #pragma unroll
    for (int v = 0; v < 8; ++v) {
      const int bout = mtile * 16 + v + half * 8;
      out[(((size_t)bout * N_IN + n) * N_OUT + m) * D_OUT + l16] = c[v];
    }
  }
}

extern "C" void kernel_launch(void* const* d_in, const int* in_sizes, int n_in,
                              void* d_out, int out_size, void* d_ws, size_t ws_size,
                              hipStream_t stream) {
  const float* in = (const float*)d_in[0];   // inputs [64,1152,8,1]
  const float* w  = (const float*)d_in[1];   // transform_mat [1,1152,10,8,16,1]
  float* out      = (float*)d_out;           // votes [64,1152,10,16,1]

  // 1152 capsules * 4 batch tiles = 4608 waves; 8 waves/block (wave32)
  dim3 block(256);
  dim3 grid(4608 / 8);
  caps_transform_wmma<<<grid, block, 0, stream>>>(in, w, out);
}